// WeightedTensorProduct_5231270166733
// MI455X (gfx1250) — compile-verified
//
#include <hip/hip_runtime.h>
#include <hip/hip_bf16.h>

typedef float v2f __attribute__((ext_vector_type(2)));
typedef float v8f __attribute__((ext_vector_type(8)));

#define C_TOT   128          // channels (reference C)
#define MOUT    16           // (L+1)^2
#define NSTEP   64           // K=256 pairs / 4 per V_WMMA_F32_16X16X4_F32
#define CH_BLK  16           // channels staged per workgroup
#define B_TILE  16           // batch tile (WMMA N)
#define BPAD    17           // LDS pad along batch dim (bank-conflict-free)
#define APACK_PER_C (NSTEP * 32 * 2)   // 4096 floats per channel, A-fragment order

// ---------------- kernel 1: zero the packed-A workspace ----------------
__global__ __launch_bounds__(256) void wtp_zero(float* __restrict__ p, int n) {
  int i = blockIdx.x * 256 + threadIdx.x;
  if (i < n) p[i] = 0.0f;
}

// ---------------- kernel 2: build Wc packed in WMMA A-fragment order ----
// apack[c][s][lane][j] = Wc[c][Mo=lane&15][k=4s+2*(lane>>4)+j],
// Wc[c][Mo][m2*16+m1] = sum_e CG[e]*weight[l_ind[e],c]  (entries unique -> plain store)
__global__ __launch_bounds__(256) void wtp_build_apack(
    const float* __restrict__ cg, const float* __restrict__ w,
    const int* __restrict__ M1,  const int* __restrict__ M2,
    const int* __restrict__ lind, const int* __restrict__ seg,
    float* __restrict__ apack, int nnz) {
  int t = blockIdx.x * 256 + threadIdx.x;
  if (t >= nnz * C_TOT) return;
  int c = t & (C_TOT - 1);
  int e = t >> 7;                       // C_TOT == 128
  int k    = M2[e] * 16 + M1[e];        // pair index, K dimension
  int s    = k >> 2;                    // WMMA step
  int kl   = k & 3;                     // K-local
  int lane = ((kl >> 1) << 4) | seg[e]; // half = kl>>1, Mo = seg
  int j    = kl & 1;
  apack[(((size_t)c * NSTEP + s) * 32 + lane) * 2 + j] =
      cg[e] * w[lind[e] * C_TOT + c];
}

// ---------------- kernel 3: main per-channel GEMM via f32 WMMA ----------
__global__ __launch_bounds__(256) void wtp_wmma_kernel(
    const float* __restrict__ x1, const float* __restrict__ x2,
    const float* __restrict__ apack, float* __restrict__ out) {
  __shared__ float s1[MOUT * CH_BLK * BPAD];
  __shared__ float s2[MOUT * CH_BLK * BPAD];

  const int tid = threadIdx.x;
  const int b0  = blockIdx.x * B_TILE;
  const int c0  = blockIdx.y * CH_BLK;

  // Stage x1/x2 tiles [16b][16m][16c] -> LDS [m][c][b(+pad)], coalesced reads.
  #pragma unroll
  for (int i = 0; i < (B_TILE * MOUT * CH_BLK) / 256; ++i) {
    int f = tid + 256 * i;
    int c = f & (CH_BLK - 1);
    int m = (f >> 4) & (MOUT - 1);
    int b = f >> 8;
    size_t g = (size_t)(b0 + b) * (MOUT * C_TOT) + m * C_TOT + (c0 + c);
    int l = (m * CH_BLK + c) * BPAD + b;
    s1[l] = x1[g];
    s2[l] = x2[g];
  }
  __syncthreads();

  const int wave = tid >> 5;
  const int lane = tid & 31;
  const int n    = lane & 15;   // WMMA N (batch within tile)
  const int h    = lane >> 4;   // lane half -> K-local {0,1} vs {2,3}

  #pragma unroll
  for (int cc = 0; cc < 2; ++cc) {
    const int cl = wave * 2 + cc;     // local channel 0..15
    const int c  = c0 + cl;

    // x2 of this lane's batch, all 16 orders
    float x2v[16];
    #pragma unroll
    for (int m = 0; m < 16; ++m) x2v[m] = s2[(m * CH_BLK + cl) * BPAD + n];
    // half-specific x1 slice: yv[2a+j] = x1[b, 4a + 2h + j]
    float yv[8];
    #pragma unroll
    for (int a = 0; a < 4; ++a)
      #pragma unroll
      for (int j = 0; j < 2; ++j)
        yv[2 * a + j] = s1[((4 * a + 2 * h + j) * CH_BLK + cl) * BPAD + n];

    v8f acc = {};
    const float* ap = apack + (size_t)c * APACK_PER_C + lane * 2;

    #pragma unroll
    for (int s = 0; s < NSTEP; ++s) {
      v2f A = *(const v2f*)(ap + s * 64);      // coalesced b64, L2-resident
      float xm2 = x2v[s >> 2];                 // m2 fixed within a K-step
      v2f Bf;
      Bf.x = xm2 * yv[2 * (s & 3) + 0];        // P[k,n] outer product, 2 muls
      Bf.y = xm2 * yv[2 * (s & 3) + 1];
      acc = __builtin_amdgcn_wmma_f32_16x16x4_f32(
          false, A, false, Bf, (short)0, acc, false, false);
    }

    // D layout: VGPR r -> Mo = r + 8*h, N = lane&15
    #pragma unroll
    for (int r = 0; r < 8; ++r) {
      int Mo = r + 8 * h;
      out[(size_t)(b0 + n) * (MOUT * C_TOT) + Mo * C_TOT + c] = acc[r];
    }
  }
}

// ---------------- launch ------------------------------------------------
extern "C" void kernel_launch(void* const* d_in, const int* in_sizes, int n_in,
                              void* d_out, int out_size, void* d_ws, size_t ws_size,
                              hipStream_t stream) {
  const float* x1 = (const float*)d_in[0];
  const float* x2 = (const float*)d_in[1];
  const float* w  = (const float*)d_in[2];
  const float* cg = (const float*)d_in[3];
  const int* M1   = (const int*)d_in[4];
  const int* M2   = (const int*)d_in[5];
  const int* li   = (const int*)d_in[6];
  const int* sg   = (const int*)d_in[7];
  float* out      = (float*)d_out;
  float* apack    = (float*)d_ws;          // C_TOT * 4096 floats = 2 MB

  const int B   = in_sizes[0] / (MOUT * C_TOT);
  const int nnz = in_sizes[3];

  const int na = C_TOT * APACK_PER_C;
  wtp_zero<<<(na + 255) / 256, 256, 0, stream>>>(apack, na);

  const int nb = nnz * C_TOT;
  wtp_build_apack<<<(nb + 255) / 256, 256, 0, stream>>>(cg, w, M1, M2, li, sg,
                                                        apack, nnz);

  dim3 grid(B / B_TILE, C_TOT / CH_BLK);   // (128, 8) for B=2048
  wtp_wmma_kernel<<<grid, 256, 0, stream>>>(x1, x2, apack, out);
}